// GCN_87273735454854
// MI455X (gfx1250) — compile-verified
//
#include <hip/hip_runtime.h>

#define N_NODES 50000
#define N_EDGES 800000
#define IN_CH   128
#define HID_CH  128
#define OUT_CH  64

typedef __attribute__((ext_vector_type(2))) float v2f;
typedef __attribute__((ext_vector_type(8))) float v8f;

// ---------------- norm computation ----------------

__global__ void k_init_deg(float* __restrict__ deg, int n) {
    int i = blockIdx.x * blockDim.x + threadIdx.x;
    if (i < n) deg[i] = 1.0f;   // self-loop contributes 1 to every node's degree
}

__global__ void k_accum_deg(const int* __restrict__ dst, float* __restrict__ deg, int ne) {
    int e = blockIdx.x * blockDim.x + threadIdx.x;
    if (e < ne) atomicAdd(&deg[dst[e]], 1.0f);
}

__global__ void k_deg_to_dinv(float* __restrict__ deg, int n) {
    int i = blockIdx.x * blockDim.x + threadIdx.x;
    if (i < n) deg[i] = rsqrtf(deg[i]);   // deg >= 1 always (self-loop)
}

__global__ void k_edge_norm(const int* __restrict__ src, const int* __restrict__ dst,
                            const float* __restrict__ dinv, float* __restrict__ norm, int ne) {
    int e = blockIdx.x * blockDim.x + threadIdx.x;
    if (e < ne) norm[e] = dinv[src[e]] * dinv[dst[e]];
}

// ---------------- dense GEMM via V_WMMA_F32_16X16X4_F32 ----------------
// One wave computes a 16x16 tile of H = act(X) @ W.  Block = (COUT/16) waves,
// covering one 16-row stripe.  X tile staged in LDS (padded stride CIN+4 for
// conflict-free reads); optional ReLU fused into the staging load (layer 2
// consumes relu(agg+b1) only here, so the separate activation pass is gone).
// W read from global (<=64KB, L2/WGP$ resident across all blocks).
//
// fp32 WMMA VGPR layouts (ISA 7.12.2):
//   A (16x4):  lanes 0-15 row M=lane  -> v0=K0, v1=K1 ; lanes 16-31 -> v0=K2, v1=K3
//   B (4x16):  VGPR v: lanes 0-15 row K=v, col=lane ; lanes 16-31 row K=v+2
//   C (16x16): VGPR v: lanes 0-15 row M=v, col=lane ; lanes 16-31 row M=v+8

template<int CIN, int COUT, bool RELU_IN>
__global__ __launch_bounds__((COUT / 16) * 32)
void k_gemm_wmma(const float* __restrict__ X, const float* __restrict__ W,
                 float* __restrict__ H) {
    constexpr int XSTRIDE = CIN + 4;            // pad: conflict-free LDS reads
    __shared__ float ldsX[16 * XSTRIDE];

    const int tid  = threadIdx.x;
    const int row0 = blockIdx.x * 16;

    // cooperative load of the 16 x CIN tile of X (ReLU fused when requested)
    for (int i = tid; i < 16 * CIN; i += blockDim.x) {
        int r = i / CIN, c = i % CIN;
        float v = X[(size_t)(row0 + r) * CIN + c];
        if (RELU_IN) v = fmaxf(v, 0.0f);
        ldsX[r * XSTRIDE + c] = v;
    }
    __syncthreads();

    const int wave = tid >> 5;
    const int lane = tid & 31;
    const int n0   = wave * 16;                 // output-column tile
    const int lm   = lane & 15;
    const int koff = (lane < 16) ? 0 : 2;

    v8f c = {};
#pragma unroll
    for (int k0 = 0; k0 < CIN; k0 += 4) {
        v2f a, b;
        a.x = ldsX[lm * XSTRIDE + k0 + koff];
        a.y = ldsX[lm * XSTRIDE + k0 + koff + 1];
        b.x = W[(size_t)(k0 + koff)     * COUT + n0 + lm];
        b.y = W[(size_t)(k0 + koff + 1) * COUT + n0 + lm];
        c = __builtin_amdgcn_wmma_f32_16x16x4_f32(
                /*neg_a=*/false, a, /*neg_b=*/false, b,
                /*c_mod=*/(short)0, c, /*reuse_a=*/false, /*reuse_b=*/false);
    }

    const int rbase = row0 + ((lane < 16) ? 0 : 8);
#pragma unroll
    for (int v = 0; v < 8; ++v)
        H[(size_t)(rbase + v) * COUT + n0 + lm] = c[v];
}

// ---------------- message passing ----------------

// Fused self-loop + bias init:  agg[i][c] = dinv[i]^2 * h[i][c] + b[c]
// (bias is additive, aggregation is a sum, so seeding with the bias is exact).
// float4-vectorized: all buffers are 16B-aligned and C % 4 == 0.
template<int C>
__global__ void k_self_init_bias(const float* __restrict__ dinv, const float* __restrict__ h,
                                 const float* __restrict__ b, float* __restrict__ agg,
                                 int total4) {
    int i = blockIdx.x * blockDim.x + threadIdx.x;
    if (i >= total4) return;
    int e0   = i * 4;
    int node = e0 / C;
    int c0   = e0 % C;
    float di = dinv[node];
    float s  = di * di;
    float4 hv = ((const float4*)h)[i];
    float4 r;
    r.x = s * hv.x + b[c0 + 0];
    r.y = s * hv.y + b[c0 + 1];
    r.z = s * hv.z + b[c0 + 2];
    r.w = s * hv.w + b[c0 + 3];
    ((float4*)agg)[i] = r;
}

// one wave32 per edge; lane l handles channels l, l+32, ...  agg is L2-resident
// (25.6MB within 192MB L2) so the f32 atomics resolve at the L2 atomic units.
template<int C>
__global__ void k_scatter(const int* __restrict__ src, const int* __restrict__ dst,
                          const float* __restrict__ norm, const float* __restrict__ h,
                          float* __restrict__ agg, int ne) {
    int t    = blockIdx.x * blockDim.x + threadIdx.x;
    int e    = t >> 5;
    int lane = t & 31;
    if (e >= ne) return;
    int   s = src[e], d = dst[e];
    float w = norm[e];
    const float* hs = h   + (size_t)s * C;
    float*       ad = agg + (size_t)d * C;
#pragma unroll
    for (int i = 0; i < C / 32; ++i)
        atomicAdd(ad + lane + i * 32, w * hs[lane + i * 32]);
}

// ---------------- launch ----------------

extern "C" void kernel_launch(void* const* d_in, const int* in_sizes, int n_in,
                              void* d_out, int out_size, void* d_ws, size_t ws_size,
                              hipStream_t stream) {
    (void)in_sizes; (void)n_in; (void)out_size; (void)ws_size;

    const float* x   = (const float*)d_in[0];
    const int*   ei  = (const int*)  d_in[1];
    const int*   src = ei;
    const int*   dst = ei + N_EDGES;
    const float* W1  = (const float*)d_in[2];
    const float* b1  = (const float*)d_in[3];
    const float* W2  = (const float*)d_in[4];
    const float* b2  = (const float*)d_in[5];
    float*       out = (float*)d_out;

    // workspace layout (all 16B-aligned: (N+E)*4 and N*128*4 are multiples of 16)
    float* dinv = (float*)d_ws;                         // N
    float* norm = dinv + N_NODES;                       // E
    float* h    = norm + N_EDGES;                       // N*128 (reused as N*64 in layer 2)
    float* agg  = h + (size_t)N_NODES * HID_CH;         // N*128

    const int T = 256;

    // ---- gcn_norm ----
    k_init_deg   <<<(N_NODES + T - 1) / T, T, 0, stream>>>(dinv, N_NODES);
    k_accum_deg  <<<(N_EDGES + T - 1) / T, T, 0, stream>>>(dst, dinv, N_EDGES);
    k_deg_to_dinv<<<(N_NODES + T - 1) / T, T, 0, stream>>>(dinv, N_NODES);
    k_edge_norm  <<<(N_EDGES + T - 1) / T, T, 0, stream>>>(src, dst, dinv, norm, N_EDGES);

    // ---- layer 1: h = x @ W1 ; agg = b1 + self + scatter(norm * h[src]) ----
    k_gemm_wmma<IN_CH, HID_CH, false><<<N_NODES / 16, (HID_CH / 16) * 32, 0, stream>>>(x, W1, h);
    {
        int total4 = N_NODES * HID_CH / 4;
        k_self_init_bias<HID_CH><<<(total4 + T - 1) / T, T, 0, stream>>>(dinv, h, b1, agg, total4);
        long long thr = (long long)N_EDGES * 32;
        k_scatter<HID_CH><<<(int)((thr + T - 1) / T), T, 0, stream>>>(src, dst, norm, h, agg, N_EDGES);
        // relu(agg) is consumed only by the next GEMM -> fused into its A-tile load
    }

    // ---- layer 2: h2 = relu(agg) @ W2 ; out = b2 + self + scatter ----
    float* h2 = h;  // reuse (N*64 <= N*128)
    k_gemm_wmma<HID_CH, OUT_CH, true><<<N_NODES / 16, (OUT_CH / 16) * 32, 0, stream>>>(agg, W2, h2);
    {
        int total4 = N_NODES * OUT_CH / 4;
        k_self_init_bias<OUT_CH><<<(total4 + T - 1) / T, T, 0, stream>>>(dinv, h2, b2, out, total4);
        long long thr = (long long)N_EDGES * 32;
        k_scatter<OUT_CH><<<(int)((thr + T - 1) / T), T, 0, stream>>>(src, dst, norm, h2, out, N_EDGES);
    }
}